// Score_45466523795916
// MI455X (gfx1250) — compile-verified
//
#include <hip/hip_runtime.h>
#include <hip/hip_bf16.h>
#include <math.h>

// Problem constants (fixed by the reference)
#define BB    4
#define NN    512
#define MM    512
#define HDIM  128
#define RROWS (BB * NN)   // 2048 rows for each projection

typedef __attribute__((ext_vector_type(2))) float v2f;
typedef __attribute__((ext_vector_type(8))) float v8f;

// ---- native tanh (gfx1250 V_TANH_F32) with safe fallbacks --------------------
#if __has_builtin(__builtin_amdgcn_tanhf)
__device__ __forceinline__ float fast_tanh(float x) { return __builtin_amdgcn_tanhf(x); }
#elif __has_builtin(__builtin_amdgcn_tanh_f32)
__device__ __forceinline__ float fast_tanh(float x) { return __builtin_amdgcn_tanh_f32(x); }
#else
__device__ __forceinline__ float fast_tanh(float x) { return tanhf(x); }
#endif

#if __has_builtin(__builtin_amdgcn_wmma_f32_16x16x4_f32)
#define HAVE_WMMA_F32 1
#endif

// =============================================================================
// Phase 1: A = first @ W1^T + b_concat   (rows 0..2047 of tile space)
//          C = second @ W2^T             (rows 2048..4095)
// One wave per 16x16 output tile, V_WMMA_F32_16X16X4_F32 over K=128 in steps of 4.
//
// f32 WMMA lane layout (ISA 7.12.2):
//   A (16x4):  lane<16: v0=K0,v1=K1 ; lane>=16: v0=K2,v1=K3 ; M = lane&15
//   B (4x16):  lane<16: v0=K0,v1=K1 ; lane>=16: v0=K2,v1=K3 ; N = lane&15
//   D (16x16): vr: M = r + 8*(lane>>4), N = lane&15
// =============================================================================
__global__ __launch_bounds__(256) void proj_wmma_kernel(
    const float* __restrict__ X1, const float* __restrict__ X2,
    const float* __restrict__ Wc, const float* __restrict__ bvec,
    float* __restrict__ Aout, float* __restrict__ Cout)
{
    const int lane      = threadIdx.x & 31;
    const int waveInBlk = threadIdx.x >> 5;
    int tile = blockIdx.x * 8 + waveInBlk;      // 0..2047
    const int which = tile >> 10;               // 0 -> A-projection, 1 -> C-projection
    tile &= 1023;
    const int row0 = (tile >> 3) * 16;          // 0..2032
    const int col0 = (tile & 7) * 16;           // 0..112

    const float* __restrict__ X   = which ? X2 : X1;
    float*       __restrict__ Out = which ? Cout : Aout;
    const int wOff = which ? HDIM : 0;          // W1 = Wc[:, :H], W2 = Wc[:, H:]

    const int half = lane >> 4;                 // 0 or 1
    const int l16  = lane & 15;

    v8f acc = {0.f, 0.f, 0.f, 0.f, 0.f, 0.f, 0.f, 0.f};

#if defined(HAVE_WMMA_F32)
    // per-lane base pointers; K slice offset 2*half selects K{0,1} vs K{2,3}
    const float* xr = X  + (size_t)(row0 + l16) * HDIM       + 2 * half;
    const float* wr = Wc + (size_t)(col0 + l16) * (2 * HDIM) + wOff + 2 * half;
    #pragma unroll 4
    for (int h0 = 0; h0 < HDIM; h0 += 4) {
        v2f a; a.x = xr[h0]; a.y = xr[h0 + 1];      // A[M=l16, K=h0+2h+{0,1}]
        v2f b; b.x = wr[h0]; b.y = wr[h0 + 1];      // B[K, N=l16] = W[col, h]
        acc = __builtin_amdgcn_wmma_f32_16x16x4_f32(
                  false, a, false, b, (short)0, acc, false, false);
    }
#else
    // scalar fallback (same D layout)
    #pragma unroll
    for (int r = 0; r < 8; ++r) {
        const float* xr = X  + (size_t)(row0 + r + 8 * half) * HDIM;
        const float* wr = Wc + (size_t)(col0 + l16) * (2 * HDIM) + wOff;
        float s = 0.f;
        for (int h = 0; h < HDIM; ++h) s = fmaf(xr[h], wr[h], s);
        acc[r] = s;
    }
#endif

    // fold b_concat into the A projection only: tanh(a + c + b) = tanh(a' + c)
    const float bias = which ? 0.f : bvec[col0 + l16];
    #pragma unroll
    for (int r = 0; r < 8; ++r) {
        Out[(size_t)(row0 + r + 8 * half) * HDIM + col0 + l16] = acc[r] + bias;
    }
}

// =============================================================================
// Phase 2: out[b,n,m] = sum_k w[k]*tanh(A[b,n,k] + C[b,m,k]) + b_single
// 32x32 output tile per 256-thread block; each thread owns 1 n x 4 m.
// LDS row stride 129: c-reads across the 8 m-groups hit 8 distinct banks
// (delta = 4*129 mod 64 = 4); a/w reads are same-address broadcasts.
// =============================================================================
#define TN 32
#define TM 32

__global__ __launch_bounds__(256) void score_tanh_kernel(
    const float* __restrict__ Abuf, const float* __restrict__ Cbuf,
    const float* __restrict__ Wsingle, const float* __restrict__ bsingle,
    float* __restrict__ Out)
{
    __shared__ float sA[TN][HDIM + 1];
    __shared__ float sC[TM][HDIM + 1];
    __shared__ float sW[HDIM];

    const int tid = threadIdx.x;
    const int b   = blockIdx.z;
    const int n0  = blockIdx.y * TN;
    const int m0  = blockIdx.x * TM;

    const float* Arows = Abuf + ((size_t)b * NN + n0) * HDIM;
    const float* Crows = Cbuf + ((size_t)b * MM + m0) * HDIM;

    // stage tiles: 32 rows x 128 floats each = 1024 float4 loads, 4 per thread
    #pragma unroll
    for (int i = 0; i < 4; ++i) {
        const int idx = tid + i * 256;       // 0..1023
        const int r   = idx >> 5;            // row 0..31
        const int c   = (idx & 31) * 4;      // col 0..124
        const float4 va = *(const float4*)(Arows + r * HDIM + c);
        sA[r][c + 0] = va.x; sA[r][c + 1] = va.y;
        sA[r][c + 2] = va.z; sA[r][c + 3] = va.w;
        const float4 vc = *(const float4*)(Crows + r * HDIM + c);
        sC[r][c + 0] = vc.x; sC[r][c + 1] = vc.y;
        sC[r][c + 2] = vc.z; sC[r][c + 3] = vc.w;
    }
    if (tid < HDIM) sW[tid] = Wsingle[tid];
    __syncthreads();

    const int nl = tid >> 3;         // local n: 0..31
    const int mg = (tid & 7) * 4;    // local m base: 0,4,...,28

    float acc0 = 0.f, acc1 = 0.f, acc2 = 0.f, acc3 = 0.f;
    #pragma unroll 4
    for (int k = 0; k < HDIM; ++k) {
        const float a  = sA[nl][k];
        const float wk = sW[k];
        acc0 = fmaf(wk, fast_tanh(a + sC[mg + 0][k]), acc0);
        acc1 = fmaf(wk, fast_tanh(a + sC[mg + 1][k]), acc1);
        acc2 = fmaf(wk, fast_tanh(a + sC[mg + 2][k]), acc2);
        acc3 = fmaf(wk, fast_tanh(a + sC[mg + 3][k]), acc3);
    }

    const float bs = bsingle[0];
    float* o = Out + ((size_t)b * NN + (n0 + nl)) * MM + m0 + mg;
    o[0] = acc0 + bs; o[1] = acc1 + bs; o[2] = acc2 + bs; o[3] = acc3 + bs;
}

// =============================================================================
extern "C" void kernel_launch(void* const* d_in, const int* in_sizes, int n_in,
                              void* d_out, int out_size, void* d_ws, size_t ws_size,
                              hipStream_t stream) {
    const float* first  = (const float*)d_in[0];  // (B,N,H)
    const float* second = (const float*)d_in[1];  // (B,M,H)
    const float* Wc     = (const float*)d_in[2];  // (H, 2H)
    const float* bc     = (const float*)d_in[3];  // (H,)
    const float* Ws     = (const float*)d_in[4];  // (1, H)
    const float* bs     = (const float*)d_in[5];  // (1,)
    float* out = (float*)d_out;                   // (B,N,M)

    float* Abuf = (float*)d_ws;                   // 2048*128 floats = 1 MB
    float* Cbuf = Abuf + (size_t)RROWS * HDIM;    // 1 MB

    // Phase 1: 2048 wave-tiles (A + C projections), 8 waves per block
    proj_wmma_kernel<<<256, 256, 0, stream>>>(first, second, Wc, bc, Abuf, Cbuf);

    // Phase 2: tanh-score
    dim3 grid(MM / TM, NN / TN, BB);
    score_tanh_kernel<<<grid, 256, 0, stream>>>(Abuf, Cbuf, Ws, bs, out);
}